// GraphSAGE_74586402062468
// MI455X (gfx1250) — compile-verified
//
#include <hip/hip_runtime.h>

typedef __attribute__((ext_vector_type(2))) float v2f;
typedef __attribute__((ext_vector_type(8))) float v8f;

// ---------------------------------------------------------------------------
// Zero-fill (float granularity, grid-stride)
// ---------------------------------------------------------------------------
__global__ void sage_zero(float* __restrict__ p, long n) {
    long i = (long)blockIdx.x * blockDim.x + threadIdx.x;
    long stride = (long)gridDim.x * blockDim.x;
    for (; i < n; i += stride) p[i] = 0.0f;
}

// ---------------------------------------------------------------------------
// Edge aggregation: one wave per edge, 32 lanes x float4 = 128 channels.
// sum[dst][:] += x[src][:]; optional deg[dst] += 1.
// x (51.2 MB) and sum (51.2 MB) are L2-resident on MI455X (192 MB L2).
// ---------------------------------------------------------------------------
__global__ __launch_bounds__(256)
void sage_aggregate(const float* __restrict__ xin,
                    const long long* __restrict__ eidx,
                    float* __restrict__ sum,
                    float* __restrict__ deg,
                    int nE, int with_deg) {
    const int lane   = threadIdx.x & 31;
    const int gwave  = ((int)blockIdx.x * 256 + (int)threadIdx.x) >> 5;
    const int nwaves = (int)gridDim.x * 8;
    for (int e = gwave; e < nE; e += nwaves) {
        const long long src = eidx[e];
        const long long dst = eidx[(long long)nE + e];
        const float4 v = *(const float4*)(xin + src * 128 + (lane << 2));
        float* s = sum + dst * 128 + (lane << 2);
        __hip_atomic_fetch_add(s + 0, v.x, __ATOMIC_RELAXED, __HIP_MEMORY_SCOPE_AGENT);
        __hip_atomic_fetch_add(s + 1, v.y, __ATOMIC_RELAXED, __HIP_MEMORY_SCOPE_AGENT);
        __hip_atomic_fetch_add(s + 2, v.z, __ATOMIC_RELAXED, __HIP_MEMORY_SCOPE_AGENT);
        __hip_atomic_fetch_add(s + 3, v.w, __ATOMIC_RELAXED, __HIP_MEMORY_SCOPE_AGENT);
        if (with_deg && lane == 0)
            __hip_atomic_fetch_add(deg + dst, 1.0f, __ATOMIC_RELAXED, __HIP_MEMORY_SCOPE_AGENT);
    }
}

// ---------------------------------------------------------------------------
// Fused SAGE layer: out = [mean | x] @ [Wl^T ; Wr^T] + b  (+ optional ReLU)
// Block = 256 threads = 8 waves, one 16-row node tile per block.
// Wave w computes the 16x16 output tile at columns [16w, 16w+16) using
// V_WMMA_F32_16X16X4_F32 (K=4 per issue, 64 issues per wave).
//
// VGPR layouts (CDNA5 ISA 7.12.2, wave32):
//   A 16x4 f32 : lane l holds row M=l%16, K = 2*(l/16) + {0,1}   -> v2f
//   B 4x16 f32 : lane l holds col N=l%16, K = 2*(l/16) + {0,1}   -> v2f
//   C/D 16x16  : lane l col N=l%16; VGPR i row M = i + 8*(l/16)  -> v8f
// B[k][n] = W[n][k]: W rows are contiguous in k -> float2 global loads.
// ---------------------------------------------------------------------------
#define LDSP 132  // padded row stride (words): 132%64==4 -> conflict-free
__global__ __launch_bounds__(256)
void sage_layer(const float* __restrict__ sum, const float* __restrict__ deg,
                const float* __restrict__ xin,
                const float* __restrict__ Wl, const float* __restrict__ bl,
                const float* __restrict__ Wr,
                float* __restrict__ out, int nNodes, int relu_flag) {
    __shared__ alignas(16) float lds_m[16][LDSP];  // mean tile
    __shared__ alignas(16) float lds_x[16][LDSP];  // self-feature tile

    const int r0 = (int)blockIdx.x * 16;
    const int t  = threadIdx.x;

    // --- stage 16x128 mean tile (sum/deg) and 16x128 x tile into LDS ---
    #pragma unroll
    for (int i = 0; i < 2; ++i) {
        const int idx = t + i * 256;           // 0..511, float4 granule
        const int row = idx >> 5;              // 32 float4 per row
        const int col = (idx & 31) << 2;
        int r = r0 + row;
        if (r >= nNodes) r = nNodes - 1;       // clamp (N%16==0 -> no-op)
        const float d   = deg[r];
        const float inv = 1.0f / fmaxf(d, 1.0f);
        float4 s4 = *(const float4*)(sum + (long)r * 128 + col);
        s4.x *= inv; s4.y *= inv; s4.z *= inv; s4.w *= inv;
        *(float4*)&lds_m[row][col] = s4;
        *(float4*)&lds_x[row][col] = *(const float4*)(xin + (long)r * 128 + col);
    }
    __syncthreads();

    // --- WMMA compute: no divergence, EXEC all-1s ---
    const int lane = t & 31;
    const int w    = t >> 5;
    const int m    = lane & 15;
    const int kk   = (lane >> 4) << 1;
    const int n    = (w << 4) + (lane & 15);

    v8f acc = {0.f, 0.f, 0.f, 0.f, 0.f, 0.f, 0.f, 0.f};
    const float* wl_row = Wl + (long)n * 128;
    const float* wr_row = Wr + (long)n * 128;

    #pragma unroll 8
    for (int k = 0; k < 128; k += 4) {
        v2f a = *(const v2f*)&lds_m[m][k + kk];
        v2f b = *(const v2f*)(wl_row + k + kk);
        acc = __builtin_amdgcn_wmma_f32_16x16x4_f32(false, a, false, b,
                                                    (short)0, acc, false, false);
    }
    #pragma unroll 8
    for (int k = 0; k < 128; k += 4) {
        v2f a = *(const v2f*)&lds_x[m][k + kk];
        v2f b = *(const v2f*)(wr_row + k + kk);
        acc = __builtin_amdgcn_wmma_f32_16x16x4_f32(false, a, false, b,
                                                    (short)0, acc, false, false);
    }

    // --- epilogue: bias (+ ReLU), scatter D tile ---
    const float bias  = bl[n];
    const int   mbase = r0 + ((lane >> 4) << 3);
    #pragma unroll
    for (int i = 0; i < 8; ++i) {
        float v = acc[i] + bias;
        if (relu_flag) v = fmaxf(v, 0.0f);
        const int r = mbase + i;
        if (r < nNodes) out[(long)r * 128 + n] = v;
    }
}

// ---------------------------------------------------------------------------
// Launch: zero -> aggregate(x) -> layer1(relu) -> zero -> aggregate(h) -> layer2
// Workspace: sum[N*128] | deg[N] | h[N*128]
// ---------------------------------------------------------------------------
extern "C" void kernel_launch(void* const* d_in, const int* in_sizes, int n_in,
                              void* d_out, int out_size, void* d_ws, size_t ws_size,
                              hipStream_t stream) {
    const float*     x    = (const float*)d_in[0];
    const long long* eidx = (const long long*)d_in[1];
    const float*     W1l  = (const float*)d_in[2];
    const float*     b1   = (const float*)d_in[3];
    const float*     W1r  = (const float*)d_in[4];
    const float*     W2l  = (const float*)d_in[5];
    const float*     b2   = (const float*)d_in[6];
    const float*     W2r  = (const float*)d_in[7];
    float*           outp = (float*)d_out;

    const int n = in_sizes[0] / 128;
    const int e = in_sizes[1] / 2;

    float* sum = (float*)d_ws;
    float* deg = sum + (size_t)n * 128;
    float* h   = deg + (size_t)n;

    const int zeroBlocks = 2048, aggBlocks = 6144;
    const int tileBlocks = (n + 15) / 16;

    // layer 1
    sage_zero<<<zeroBlocks, 256, 0, stream>>>(sum, (long)n * 129);  // sum + deg
    sage_aggregate<<<aggBlocks, 256, 0, stream>>>(x, eidx, sum, deg, e, 1);
    sage_layer<<<tileBlocks, 256, 0, stream>>>(sum, deg, x, W1l, b1, W1r, h, n, 1);

    // layer 2 (same graph -> reuse deg; reuse sum buffer)
    sage_zero<<<zeroBlocks, 256, 0, stream>>>(sum, (long)n * 128);
    sage_aggregate<<<aggBlocks, 256, 0, stream>>>(h, eidx, sum, nullptr, e, 0);
    sage_layer<<<tileBlocks, 256, 0, stream>>>(sum, deg, h, W2l, b2, W2r, outp, n, 0);
}